// GCN_69269232549994
// MI455X (gfx1250) — compile-verified
//
#include <hip/hip_runtime.h>

typedef __attribute__((ext_vector_type(2))) float v2f;
typedef __attribute__((ext_vector_type(8))) float v8f;

#define THREADS 256
#define WAVES_PER_BLOCK (THREADS / 32)

// ---------------------------------------------------------------------------
// Kernel 1: deg = 1 (self loop), tmp accumulator = 0  (ws is poisoned 0xAA)
// ---------------------------------------------------------------------------
__global__ void k_init(float* __restrict__ deg, float* __restrict__ tmp, int N) {
    int i = blockIdx.x * blockDim.x + threadIdx.x;
    if (i < N) {
        deg[i] = 1.0f;
        tmp[3 * (size_t)i + 0] = 0.0f;
        tmp[3 * (size_t)i + 1] = 0.0f;
        tmp[3 * (size_t)i + 2] = 0.0f;
    }
}

// ---------------------------------------------------------------------------
// Kernel 2: degree count over dst (edge_index row 1).  2 MB target -> L2.
// ---------------------------------------------------------------------------
__global__ void k_deg(const long long* __restrict__ dst, float* __restrict__ deg, int E) {
    int e = blockIdx.x * blockDim.x + threadIdx.x;
    if (e < E) atomicAdd(&deg[(int)dst[e]], 1.0f);
}

// ---------------------------------------------------------------------------
// Kernel 3: dinv = rsqrt(deg) in place (deg >= 1 always due to self loop)
// ---------------------------------------------------------------------------
__global__ void k_dinv(float* __restrict__ deg, int N) {
    int i = blockIdx.x * blockDim.x + threadIdx.x;
    if (i < N) deg[i] = rsqrtf(deg[i]);
}

// ---------------------------------------------------------------------------
// Kernel 4: xws[i,:] = dinv[i] * (x[i,:] @ W.T)   via V_WMMA_F32_16X16X4_F32
// One wave32 computes a 16-node tile: 32 chained K=4 WMMAs over K=128.
// A (16x4 f32): lane<16 -> row=lane, {K+0,K+1}; lane>=16 -> row=lane-16, {K+2,K+3}
// B (4x16 f32): mirrored; W zero-padded to 16 rows in LDS so EXEC stays all-1.
// C (16x16 f32): VGPR v, lane l -> row v+8*(l>=16), col l%16 (cols 0..2 valid).
// ---------------------------------------------------------------------------
__global__ void __launch_bounds__(THREADS)
k_xw_wmma(const float* __restrict__ x, const float* __restrict__ W,
          const float* __restrict__ dinv, float* __restrict__ xws,
          int N, int ntiles) {
    __shared__ float sW[16 * 128];  // rows 0..2 = gcn_W, rows 3..15 = 0
    for (int t = threadIdx.x; t < 16 * 128; t += THREADS)
        sW[t] = (t < 3 * 128) ? W[t] : 0.0f;
    __syncthreads();

    const int wave = threadIdx.x >> 5;
    const int lane = threadIdx.x & 31;
    const int tile = blockIdx.x * WAVES_PER_BLOCK + wave;
    if (tile >= ntiles) return;  // wave-uniform exit: EXEC stays all-1s below

    const int base = tile * 16;
    const int mrow = lane & 15;   // A-row within tile / B-column index
    const int hi   = lane >> 4;   // 0: K+0,K+1   1: K+2,K+3

    int arow = base + mrow;
    if (arow >= N) arow = N - 1;  // clamp (dupes are discarded at store)
    const float* xrow = x + (size_t)arow * 128 + 2 * hi;
    const float* wrow = sW + (size_t)mrow * 128 + 2 * hi;

    v8f c = {0.f, 0.f, 0.f, 0.f, 0.f, 0.f, 0.f, 0.f};
#pragma unroll
    for (int kk = 0; kk < 32; ++kk) {
        v2f a, b;
        a.x = xrow[4 * kk + 0];
        a.y = xrow[4 * kk + 1];
        b.x = wrow[4 * kk + 0];
        b.y = wrow[4 * kk + 1];
        // D = A(16x4) * B(4x16) + C, f32 accumulate
        c = __builtin_amdgcn_wmma_f32_16x16x4_f32(
                /*neg_a=*/false, a, /*neg_b=*/false, b,
                /*c_mod=*/(short)0, c, /*reuse_a=*/false, /*reuse_b=*/false);
    }

    // scatter out: lane holds col=mrow of rows base + v + 8*hi, v=0..7
#pragma unroll
    for (int v = 0; v < 8; ++v) {
        int row = base + v + 8 * hi;
        if (mrow < 3 && row < N)
            xws[(size_t)row * 3 + mrow] = c[v] * dinv[row];
    }
}

// ---------------------------------------------------------------------------
// Kernel 5: per-edge gather xws[src] (12B, L2-resident) + 3 atomic adds
// ---------------------------------------------------------------------------
__global__ void k_scatter(const long long* __restrict__ src,
                          const long long* __restrict__ dst,
                          const float* __restrict__ xws,
                          float* __restrict__ tmp, int E) {
    int e = blockIdx.x * blockDim.x + threadIdx.x;
    if (e >= E) return;
    int s = (int)src[e];
    int d = (int)dst[e];
    float m0 = xws[(size_t)s * 3 + 0];
    float m1 = xws[(size_t)s * 3 + 1];
    float m2 = xws[(size_t)s * 3 + 2];
    atomicAdd(&tmp[(size_t)d * 3 + 0], m0);
    atomicAdd(&tmp[(size_t)d * 3 + 1], m1);
    atomicAdd(&tmp[(size_t)d * 3 + 2], m2);
}

// ---------------------------------------------------------------------------
// Kernel 6: h = relu(dinv*(tmp + xws) + gcn_b);  z = h @ out_W.T + out_b
// (self-loop term dinv[d]^2 * xw[d] == dinv[d] * xws[d])
// ---------------------------------------------------------------------------
__global__ void k_final(const float* __restrict__ tmp, const float* __restrict__ xws,
                        const float* __restrict__ dinv,
                        const float* __restrict__ gcn_b,
                        const float* __restrict__ out_W,
                        const float* __restrict__ out_b,
                        float* __restrict__ h_out, float* __restrict__ z_out, int N) {
    int i = blockIdx.x * blockDim.x + threadIdx.x;
    if (i >= N) return;
    float di = dinv[i];
    float h0 = fmaxf(di * (tmp[(size_t)i * 3 + 0] + xws[(size_t)i * 3 + 0]) + gcn_b[0], 0.0f);
    float h1 = fmaxf(di * (tmp[(size_t)i * 3 + 1] + xws[(size_t)i * 3 + 1]) + gcn_b[1], 0.0f);
    float h2 = fmaxf(di * (tmp[(size_t)i * 3 + 2] + xws[(size_t)i * 3 + 2]) + gcn_b[2], 0.0f);
    h_out[(size_t)i * 3 + 0] = h0;
    h_out[(size_t)i * 3 + 1] = h1;
    h_out[(size_t)i * 3 + 2] = h2;
#pragma unroll
    for (int j = 0; j < 4; ++j) {
        float z = fmaf(h0, out_W[j * 3 + 0],
                  fmaf(h1, out_W[j * 3 + 1],
                  fmaf(h2, out_W[j * 3 + 2], out_b[j])));
        z_out[(size_t)i * 4 + j] = z;
    }
}

// ---------------------------------------------------------------------------
extern "C" void kernel_launch(void* const* d_in, const int* in_sizes, int n_in,
                              void* d_out, int out_size, void* d_ws, size_t ws_size,
                              hipStream_t stream) {
    const float*     x     = (const float*)d_in[0];
    const long long* ei    = (const long long*)d_in[1];  // int64 [2, E]
    const float*     gcn_W = (const float*)d_in[2];
    const float*     gcn_b = (const float*)d_in[3];
    const float*     out_W = (const float*)d_in[4];
    const float*     out_b = (const float*)d_in[5];

    const int N = in_sizes[0] / 128;
    const int E = in_sizes[1] / 2;
    const long long* src = ei;       // edge_index[0]
    const long long* dst = ei + E;   // edge_index[1]

    // workspace: deg/dinv [N] | xws [3N] | tmp [3N]   (~14 MB, L2-resident)
    float* deg = (float*)d_ws;
    float* xws = deg + (size_t)N;
    float* tmp = deg + 4 * (size_t)N;

    float* h_out = (float*)d_out;                 // [N,3]
    float* z_out = h_out + 3 * (size_t)N;         // [N,4]

    const int nblkN = (N + THREADS - 1) / THREADS;
    const int nblkE = (E + THREADS - 1) / THREADS;
    const int ntiles = (N + 15) / 16;
    const int nblkT = (ntiles + WAVES_PER_BLOCK - 1) / WAVES_PER_BLOCK;

    k_init   <<<nblkN, THREADS, 0, stream>>>(deg, tmp, N);
    k_deg    <<<nblkE, THREADS, 0, stream>>>(dst, deg, E);
    k_dinv   <<<nblkN, THREADS, 0, stream>>>(deg, N);
    k_xw_wmma<<<nblkT, THREADS, 0, stream>>>(x, gcn_W, deg, xws, N, ntiles);
    k_scatter<<<nblkE, THREADS, 0, stream>>>(src, dst, xws, tmp, E);
    k_final  <<<nblkN, THREADS, 0, stream>>>(tmp, xws, deg, gcn_b, out_W, out_b,
                                             h_out, z_out, N);
}